// MultiBoxLoss_15891378995708
// MI455X (gfx1250) — compile-verified
//
#include <hip/hip_runtime.h>
#include <math.h>

#define NUMB 64
#define PP   8732
#define CC   81
#define KK   16
#define TILE 96            // priors per TDM tile: 96*81*4 = 31,104 B per buffer
#define THRESH 0.5f
#define NEGPOS 3

typedef unsigned int  u32x4 __attribute__((ext_vector_type(4)));
typedef int           i32x8 __attribute__((ext_vector_type(8)));
typedef int           i32x4 __attribute__((ext_vector_type(4)));

// ---------------------------------------------------------------------------
// TDM: contiguous 1-D DMA of `nelem` f32 from global memory into LDS.
// D# group0: count=1, lds_addr, global_addr[56:0], type=2.
// D# group1: data_size=4B, tensor_dim0=nelem, tensor_dim1=1, tile_dim0=nelem,
//            tile_dim1/2=0 (1-D), tensor_dim0_stride=nelem.
// Groups 2/3 zero (<=2D tensor); 5th (extra) group zero.  Issued by one wave;
// EXEC is ignored by TDM; completion tracked with TENSORcnt.
// ---------------------------------------------------------------------------
__device__ __forceinline__ void tdm_load_f32(const float* gsrc, unsigned lds_byte,
                                             unsigned nelem) {
  unsigned long long ga = (unsigned long long)(uintptr_t)gsrc;
  u32x4 g0;
  g0.x = 1u;                                    // count = 1 (valid descriptor)
  g0.y = lds_byte;                              // LDS byte address
  g0.z = (unsigned)ga;                          // global_addr[31:0]
  g0.w = (unsigned)(ga >> 32) | (2u << 30);     // global_addr[56:32] | type=2
  i32x8 g1;
  g1[0] = 0x00020000;                                       // data_size=2 -> 4 bytes
  g1[1] = (int)((nelem & 0xFFFFu) << 16);                   // tensor_dim0[15:0]
  g1[2] = (int)(((nelem >> 16) & 0xFFFFu) | (1u << 16));    // dim0 hi | tensor_dim1=1
  g1[3] = (int)((nelem & 0xFFFFu) << 16);                   // tile_dim0 (<=10368)
  g1[4] = 0;                                                // tile_dim1/2 = 0
  g1[5] = (int)nelem;                                       // tensor_dim0_stride lo
  g1[6] = 0;
  g1[7] = 0;
  i32x4 g2 = {0, 0, 0, 0};
  i32x4 g3 = {0, 0, 0, 0};
  i32x8 g4 = {0, 0, 0, 0, 0, 0, 0, 0};
  __builtin_amdgcn_tensor_load_to_lds(g0, g1, g2, g3, g4, 0);
}

__device__ __forceinline__ float sl1(float d) {
  float a = fabsf(d);
  return (a < 1.0f) ? 0.5f * d * d : (a - 0.5f);
}

// ---------------------------------------------------------------------------
// Kernel 0: zero the accumulators (lossl, lossc, nposTotal).
// ---------------------------------------------------------------------------
__global__ void k_init(float* acc) {
  acc[0] = 0.0f;
  acc[1] = 0.0f;
  ((int*)acc)[2] = 0;
}

// ---------------------------------------------------------------------------
// Kernel 1: per-image matching + encode + smooth-L1.  One WG (256 thr)/image.
// ---------------------------------------------------------------------------
__global__ void k_match(const float* __restrict__ loc, const float* __restrict__ priors,
                        const float* __restrict__ targets, int* __restrict__ conf_t,
                        int* __restrict__ numPos, float* __restrict__ acc) {
  __shared__ float         s_btOv[PP];
  __shared__ unsigned char s_btK[PP];
  __shared__ float s_tx1[KK], s_ty1[KK], s_tx2[KK], s_ty2[KK], s_lab[KK], s_area[KK];
  __shared__ float s_redO[256];
  __shared__ int   s_redI[256];
  __shared__ int   s_redN[256];
  __shared__ float s_bpOv[KK];
  __shared__ int   s_bpIdx[KK];

  const int b = blockIdx.x, tid = threadIdx.x;
  if (tid < KK) {
    const float* t = targets + (size_t)(b * KK + tid) * 6;
    s_lab[tid] = t[1];
    s_tx1[tid] = t[2]; s_ty1[tid] = t[3]; s_tx2[tid] = t[4]; s_ty2[tid] = t[5];
    s_area[tid] = (t[4] - t[2]) * (t[5] - t[3]);
  }
  __syncthreads();

  float bOv[KK]; int bIdx[KK];
#pragma unroll
  for (int k = 0; k < KK; k++) { bOv[k] = -1.0f; bIdx[k] = 0; }

  for (int p = tid; p < PP; p += 256) {
    const float* pr = priors + (size_t)p * 4;
    float px1 = pr[0], py1 = pr[1], px2 = pr[2], py2 = pr[3];
    float areaB = (px2 - px1) * (py2 - py1);
    float bestO = -1.0f; int bestK = 0;
#pragma unroll
    for (int k = 0; k < KK; k++) {
      float ix1 = fmaxf(s_tx1[k], px1), iy1 = fmaxf(s_ty1[k], py1);
      float ix2 = fminf(s_tx2[k], px2), iy2 = fminf(s_ty2[k], py2);
      float iw = fmaxf(ix2 - ix1, 0.0f), ih = fmaxf(iy2 - iy1, 0.0f);
      float inter = iw * ih;
      float ov = inter / (s_area[k] + areaB - inter);
      if (ov > bestO) { bestO = ov; bestK = k; }     // first-k tie-break (k asc)
      if (ov > bOv[k]) { bOv[k] = ov; bIdx[k] = p; } // first-p tie-break (p asc)
    }
    s_btOv[p] = bestO;
    s_btK[p]  = (unsigned char)bestK;
  }
  __syncthreads();

  // cross-thread argmax per truth (lowest prior index wins on ties)
  for (int k = 0; k < KK; k++) {
    s_redO[tid] = bOv[k]; s_redI[tid] = bIdx[k];
    __syncthreads();
    for (int s = 128; s > 0; s >>= 1) {
      if (tid < s) {
        float o2 = s_redO[tid + s]; int i2 = s_redI[tid + s];
        if (o2 > s_redO[tid] || (o2 == s_redO[tid] && i2 < s_redI[tid])) {
          s_redO[tid] = o2; s_redI[tid] = i2;
        }
      }
      __syncthreads();
    }
    if (tid == 0) { s_bpOv[k] = s_redO[0]; s_bpIdx[k] = s_redI[0]; }
    __syncthreads();
  }
  // force-assign scatter, last write wins (k ascending like the reference loop)
  if (tid == 0) {
    for (int k = 0; k < KK; k++) {
      int p = s_bpIdx[k];
      s_btK[p]  = (unsigned char)k;
      s_btOv[p] = s_bpOv[k];
    }
  }
  __syncthreads();

  float lossl = 0.0f; int npos = 0;
  for (int p = tid; p < PP; p += 256) {
    int k = s_btK[p];
    float ov = s_btOv[p];
    int cls = (ov < THRESH) ? 0 : (int)s_lab[k];
    conf_t[(size_t)b * PP + p] = cls;
    if (cls > 0) {
      npos++;
      const float* pr = priors + (size_t)p * 4;
      float pcx = 0.5f * (pr[0] + pr[2]), pcy = 0.5f * (pr[1] + pr[3]);
      float pw = pr[2] - pr[0], ph = pr[3] - pr[1];
      float mx1 = s_tx1[k], my1 = s_ty1[k], mx2 = s_tx2[k], my2 = s_ty2[k];
      float gx = (0.5f * (mx1 + mx2) - pcx) / (0.1f * pw);
      float gy = (0.5f * (my1 + my2) - pcy) / (0.1f * ph);
      float gw = logf((mx2 - mx1) / pw) / 0.2f;
      float gh = logf((my2 - my1) / ph) / 0.2f;
      const float* lp = loc + ((size_t)b * PP + p) * 4;
      lossl += sl1(lp[0] - gx) + sl1(lp[1] - gy) + sl1(lp[2] - gw) + sl1(lp[3] - gh);
    }
  }
  s_redO[tid] = lossl; s_redN[tid] = npos;
  __syncthreads();
  for (int s = 128; s > 0; s >>= 1) {
    if (tid < s) { s_redO[tid] += s_redO[tid + s]; s_redN[tid] += s_redN[tid + s]; }
    __syncthreads();
  }
  if (tid == 0) {
    atomicAdd(&acc[0], s_redO[0]);
    numPos[b] = s_redN[0];
    atomicAdd(((int*)acc) + 2, s_redN[0]);
  }
}

// ---------------------------------------------------------------------------
// Kernel 2: CE per prior.  TDM double-buffers 96-prior tiles of conf into LDS;
// one thread per prior reduces its contiguous 81-class row from LDS
// (stride 81 mod 64 = 17 -> bank-conflict-free within a wave32).
// ---------------------------------------------------------------------------
__global__ void k_ce(const float* __restrict__ conf, const int* __restrict__ conf_t,
                     float* __restrict__ ce) {
  __shared__ float buf[2][TILE * CC];   // 2 * 31,104 B = 62,208 B
  const int b = blockIdx.y, tid = threadIdx.x;
  const int nTiles = (PP + TILE - 1) / TILE;  // 91
  const int t0 = blockIdx.x;

  if (tid == 0 && t0 < nTiles) {
    int p0 = t0 * TILE;
    int cnt = min(TILE, PP - p0);
    tdm_load_f32(conf + ((size_t)b * PP + p0) * CC,
                 (unsigned)(uintptr_t)&buf[0][0], (unsigned)(cnt * CC));
  }
  int par = 0;
  for (int t = t0; t < nTiles; t += gridDim.x) {
    int tn = t + gridDim.x;
    if (tid == 0) {
      __builtin_amdgcn_s_wait_tensorcnt(0);   // current buffer landed
      if (tn < nTiles) {                      // prefetch next tile into other buffer
        int p0n = tn * TILE;
        int cntn = min(TILE, PP - p0n);
        tdm_load_f32(conf + ((size_t)b * PP + p0n) * CC,
                     (unsigned)(uintptr_t)&buf[par ^ 1][0], (unsigned)(cntn * CC));
      }
    }
    __syncthreads();

    int p0 = t * TILE;
    int cnt = min(TILE, PP - p0);
    if (tid < cnt) {
      const float* row = &buf[par][tid * CC];
      float m = row[0];
      for (int c = 1; c < CC; c++) m = fmaxf(m, row[c]);
      float s = 0.0f;
      for (int c = 0; c < CC; c++) s += expf(row[c] - m);
      int ctv = conf_t[(size_t)b * PP + p0 + tid];
      ce[(size_t)b * PP + p0 + tid] = m + logf(s) - row[ctv];
    }
    __syncthreads();
    par ^= 1;
  }
}

// ---------------------------------------------------------------------------
// Kernel 3: hard-negative mining via exact radix-select (matches stable
// double-argsort: boundary ties all share the same value, so the CE sum is
// identity-independent).  One WG (256 thr)/image.
// ---------------------------------------------------------------------------
__global__ void k_mine(const int* __restrict__ conf_t, const float* __restrict__ ce,
                       const int* __restrict__ numPos, float* __restrict__ acc) {
  __shared__ unsigned s_keys[PP];   // 34,928 B
  __shared__ int      s_hist[256];
  __shared__ unsigned s_prefix;
  __shared__ int      s_R;
  __shared__ float    s_redF[256];
  __shared__ int      s_redI[256];

  const int b = blockIdx.x, tid = threadIdx.x;
  const int*   ct = conf_t + (size_t)b * PP;
  const float* cp = ce + (size_t)b * PP;

  for (int i = tid; i < PP; i += 256) {
    float lm = (ct[i] > 0) ? 0.0f : cp[i];
    unsigned u = __float_as_uint(lm);
    u ^= (u >> 31) ? 0xFFFFFFFFu : 0x80000000u;   // order-preserving key
    s_keys[i] = u;
  }
  int np = numPos[b];
  int nn = min(NEGPOS * np, PP - 1);
  if (tid == 0) { s_prefix = 0u; s_R = nn; }
  __syncthreads();

  if (nn > 0) {
    for (int shift = 24; shift >= 0; shift -= 8) {
      s_hist[tid] = 0;
      __syncthreads();
      unsigned pmask = (shift == 24) ? 0u : (0xFFFFFFFFu << (shift + 8));
      unsigned pref = s_prefix;
      for (int i = tid; i < PP; i += 256) {
        unsigned k = s_keys[i];
        if ((k & pmask) == pref) atomicAdd(&s_hist[(k >> shift) & 0xFF], 1);
      }
      __syncthreads();
      if (tid == 0) {
        int R = s_R, bin = 255;
        for (; bin > 0; --bin) {
          int c = s_hist[bin];
          if (R <= c) break;
          R -= c;
        }
        s_prefix = pref | ((unsigned)bin << shift);
        s_R = R;
      }
      __syncthreads();
    }
  }
  unsigned T = s_prefix;

  float s = 0.0f; int cgt = 0;
  for (int i = tid; i < PP; i += 256) {
    bool gt = (nn > 0) && (s_keys[i] > T);
    if (ct[i] > 0)      s += cp[i];   // positives always selected
    else if (gt)        s += cp[i];   // strictly-above-threshold negatives
    if (gt) cgt++;                    // rank pool includes positives too
  }
  s_redF[tid] = s; s_redI[tid] = cgt;
  __syncthreads();
  for (int st = 128; st > 0; st >>= 1) {
    if (tid < st) { s_redF[tid] += s_redF[tid + st]; s_redI[tid] += s_redI[tid + st]; }
    __syncthreads();
  }
  if (tid == 0) {
    float tot = s_redF[0];
    if (nn > 0) {
      float v = (T >= 0x80000000u) ? __uint_as_float(T ^ 0x80000000u)
                                   : __uint_as_float(~T);
      tot += (float)(nn - s_redI[0]) * v;   // boundary ties: count * value
    }
    atomicAdd(&acc[1], tot);
  }
}

// ---------------------------------------------------------------------------
// Kernel 4: finalize the two scalars.
// ---------------------------------------------------------------------------
__global__ void k_final(const float* __restrict__ acc, float* __restrict__ out) {
  float n = fmaxf((float)((const int*)acc)[2], 1.0f);
  out[0] = acc[0] / n;
  out[1] = acc[1] / n;
}

extern "C" void kernel_launch(void* const* d_in, const int* in_sizes, int n_in,
                              void* d_out, int out_size, void* d_ws, size_t ws_size,
                              hipStream_t stream) {
  (void)in_sizes; (void)n_in; (void)out_size; (void)ws_size;
  const float* loc    = (const float*)d_in[0];
  const float* conf   = (const float*)d_in[1];
  const float* priors = (const float*)d_in[2];
  const float* tgt    = (const float*)d_in[3];

  char*  w      = (char*)d_ws;
  int*   conf_t = (int*)w;                                   // 64*8732 ints
  float* ce     = (float*)(w + (size_t)NUMB * PP * 4);       // 64*8732 floats
  int*   numPos = (int*)(w + (size_t)NUMB * PP * 8);         // 64 ints
  float* acc    = (float*)(w + (size_t)NUMB * PP * 8 + 256); // lossl, lossc, nposTot
  float* out    = (float*)d_out;

  hipLaunchKernelGGL(k_init,  dim3(1),        dim3(1),    0, stream, acc);
  hipLaunchKernelGGL(k_match, dim3(NUMB),     dim3(256),  0, stream,
                     loc, priors, tgt, conf_t, numPos, acc);
  hipLaunchKernelGGL(k_ce,    dim3(16, NUMB), dim3(TILE), 0, stream,
                     conf, conf_t, ce);
  hipLaunchKernelGGL(k_mine,  dim3(NUMB),     dim3(256),  0, stream,
                     conf_t, ce, numPos, acc);
  hipLaunchKernelGGL(k_final, dim3(1),        dim3(1),    0, stream, acc, out);
}